// EfficientAttention_Mapping_7859790152142
// MI455X (gfx1250) — compile-verified
//
#include <hip/hip_runtime.h>
#include <hip/hip_bf16.h>

#define NHALF 2048
#define DKDIM 64
#define DDIM  3

typedef __attribute__((ext_vector_type(2))) float v2f;
typedef __attribute__((ext_vector_type(8))) float v8f;

// D = A(16x4 f32) x B(4x16 f32) + C(16x16 f32), wave32 distributed.
// A layout (ISA 7.12.2): lanes 0-15 -> M=lane, VGPR0=K0, VGPR1=K1;
//                        lanes 16-31 -> M=lane-16, VGPR0=K2, VGPR1=K3.
// D layout: VGPR v, lane L -> row M = v + 8*(L>=16), col N = L%16.
__device__ __forceinline__ v8f wmma16x16x4_f32(v2f a, v2f b, v8f c) {
#if defined(__AMDGCN__) && __has_builtin(__builtin_amdgcn_wmma_f32_16x16x4_f32)
  return __builtin_amdgcn_wmma_f32_16x16x4_f32(false, a, false, b, (short)0, c,
                                               false, false);
#else
  // Host-pass / non-gfx1250 placeholder (never used at runtime on gfx1250).
  c[0] += a.x * b.x + a.y * b.y;
  return c;
#endif
}

// One workgroup (256 threads = 8 waves) per (batch, half).
__global__ __launch_bounds__(256) void eff_attn_half_kernel(
    const float* __restrict__ x_all,  // [256*4096*3], block bid owns rows [bid*2048, +2048)
    const float* __restrict__ Wq,     // [3][64]
    const float* __restrict__ bq,     // [64]
    const float* __restrict__ Wk,     // [3][64]
    const float* __restrict__ bk,     // [64]
    const float* __restrict__ Wv,     // [3][3]
    const float* __restrict__ bv,     // [3]
    float* __restrict__ partial)      // [512*3]
{
  const int bid  = blockIdx.x;        // b*2 + half
  const int tid  = threadIdx.x;
  const int wave = tid >> 5;          // 0..7
  const int lane = tid & 31;
  const int l16  = lane & 15;
  const bool hi  = (lane >= 16);

  const float* xbase = x_all + (size_t)bid * NHALF * DDIM;

  __shared__ float xt[8][16][4];        // per-wave row stash for v-projection
  __shared__ float red[8][5][DKDIM];    // cross-wave reduction
  __shared__ float pred[DKDIM][4];      // final dk reduction

  // ---- B matrices (4x16 f32) + biases in registers, per the ISA B layout:
  //      VGPR0: lanes0-15 = (K=0,N=lane), lanes16-31 = (K=2,N=lane-16)
  //      VGPR1: lanes0-15 = (K=1,N=lane), lanes16-31 = (K=3 pad = 0)
  v2f Bq[4], Bk[4];
  float bqv[4], bkv[4];
#pragma unroll
  for (int t = 0; t < 4; ++t) {
    const int col = t * 16 + l16;
    Bq[t].x = hi ? Wq[2 * DKDIM + col] : Wq[0 * DKDIM + col];
    Bq[t].y = hi ? 0.0f : Wq[1 * DKDIM + col];
    Bk[t].x = hi ? Wk[2 * DKDIM + col] : Wk[0 * DKDIM + col];
    Bk[t].y = hi ? 0.0f : Wk[1 * DKDIM + col];
    bqv[t] = bq[col];
    bkv[t] = bk[col];
  }
  float wv[9], bvv[3];
#pragma unroll
  for (int i = 0; i < 9; ++i) wv[i] = Wv[i];
#pragma unroll
  for (int j = 0; j < 3; ++j) bvv[j] = bv[j];

  // Per-lane accumulators. Lane's columns are dk = 16*t + l16.
  float qacc[4] = {0.f, 0.f, 0.f, 0.f};           // sum_n softmax_dk(q)[dk]
  float sacc[4] = {0.f, 0.f, 0.f, 0.f};           // sum_n exp(k)[dk]
  float tacc[4][3] = {{0.f,0.f,0.f},{0.f,0.f,0.f},{0.f,0.f,0.f},{0.f,0.f,0.f}};

  // 2048 rows = 128 tiles of 16 rows; wave w takes tiles w, w+8, ..., w+120.
  for (int it = 0; it < 16; ++it) {
    const int tile = it * 8 + wave;
    const int row0 = tile * 16;
    const float* xr = xbase + (size_t)(row0 + l16) * DDIM;

    // Prefetch next tile's rows (global_prefetch_b8).
    if (it + 1 < 16) __builtin_prefetch(xr + 8 * 16 * DDIM, 0, 0);

    // ---- A matrix (16x4, K=3 + zero pad)
    v2f A;
    if (!hi) { A.x = xr[0]; A.y = xr[1]; }
    else     { A.x = xr[2]; A.y = 0.0f;  }

    // Stash full rows for the v-projection loop.
    if (!hi) { xt[wave][l16][0] = A.x; xt[wave][l16][1] = A.y; }
    else     { xt[wave][l16][2] = A.x; }
    __syncthreads();  // uniform across the block (same trip count per wave)

    // ================= Q path: softmax over dk, accumulate column sums ====
    v8f Eq[4];
#pragma unroll
    for (int t = 0; t < 4; ++t) {
      v8f c = {0.f, 0.f, 0.f, 0.f, 0.f, 0.f, 0.f, 0.f};
      v8f dq = wmma16x16x4_f32(A, Bq[t], c);
#pragma unroll
      for (int v = 0; v < 8; ++v) Eq[t][v] = __expf(dq[v] + bqv[t]);
    }
    // Row sums: reduce over the 16 lanes of this half + 4 column tiles.
#pragma unroll
    for (int v = 0; v < 8; ++v) {
      float r = Eq[0][v] + Eq[1][v] + Eq[2][v] + Eq[3][v];
      r += __shfl_xor(r, 1);
      r += __shfl_xor(r, 2);
      r += __shfl_xor(r, 4);
      r += __shfl_xor(r, 8);
      const float inv = 1.0f / r;
      Eq[0][v] *= inv; Eq[1][v] *= inv; Eq[2][v] *= inv; Eq[3][v] *= inv;
    }
    // Column sums of normalized q; fold the two halves together.
#pragma unroll
    for (int t = 0; t < 4; ++t) {
      float cs = Eq[t][0] + Eq[t][1] + Eq[t][2] + Eq[t][3] +
                 Eq[t][4] + Eq[t][5] + Eq[t][6] + Eq[t][7];
      cs += __shfl_xor(cs, 16);
      qacc[t] += cs;
    }

    // ================= K path: exp sums and exp-weighted v sums ===========
    v8f Ek[4];
#pragma unroll
    for (int t = 0; t < 4; ++t) {
      v8f c = {0.f, 0.f, 0.f, 0.f, 0.f, 0.f, 0.f, 0.f};
      v8f dk = wmma16x16x4_f32(A, Bk[t], c);
#pragma unroll
      for (int v = 0; v < 8; ++v) Ek[t][v] = __expf(dk[v] + bkv[t]);
    }
#pragma unroll
    for (int v = 0; v < 8; ++v) {
      const int row = v + (hi ? 8 : 0);
      const float x0 = xt[wave][row][0];
      const float x1 = xt[wave][row][1];
      const float x2 = xt[wave][row][2];
      const float v0 = x0 * wv[0] + x1 * wv[3] + x2 * wv[6] + bvv[0];
      const float v1 = x0 * wv[1] + x1 * wv[4] + x2 * wv[7] + bvv[1];
      const float v2 = x0 * wv[2] + x1 * wv[5] + x2 * wv[8] + bvv[2];
#pragma unroll
      for (int t = 0; t < 4; ++t) {
        const float e = Ek[t][v];
        sacc[t]    += e;
        tacc[t][0] += e * v0;
        tacc[t][1] += e * v1;
        tacc[t][2] += e * v2;
      }
    }
  }

  // Fold the two 16-lane halves for the k statistics.
#pragma unroll
  for (int t = 0; t < 4; ++t) {
    sacc[t]    += __shfl_xor(sacc[t], 16);
    tacc[t][0] += __shfl_xor(tacc[t][0], 16);
    tacc[t][1] += __shfl_xor(tacc[t][1], 16);
    tacc[t][2] += __shfl_xor(tacc[t][2], 16);
  }
  if (!hi) {
#pragma unroll
    for (int t = 0; t < 4; ++t) {
      const int c = t * 16 + l16;
      red[wave][0][c] = qacc[t];
      red[wave][1][c] = sacc[t];
      red[wave][2][c] = tacc[t][0];
      red[wave][3][c] = tacc[t][1];
      red[wave][4][c] = tacc[t][2];
    }
  }
  __syncthreads();

  // Cross-wave reduction: one thread per dk channel.
  if (tid < DKDIM) {
    float qs = 0.f, ss = 0.f, t0 = 0.f, t1 = 0.f, t2 = 0.f;
#pragma unroll
    for (int w = 0; w < 8; ++w) {
      qs += red[w][0][tid];
      ss += red[w][1][tid];
      t0 += red[w][2][tid];
      t1 += red[w][3][tid];
      t2 += red[w][4][tid];
    }
    // out_v = mean_n(q)[dk] * context[dk][v],  context = t / s
    const float scale = (qs * (1.0f / (float)NHALF)) / ss;
    pred[tid][0] = scale * t0;
    pred[tid][1] = scale * t1;
    pred[tid][2] = scale * t2;
  }
  __syncthreads();

  if (tid < 3) {
    float s = 0.f;
    for (int k = 0; k < DKDIM; ++k) s += pred[k][tid];
    partial[bid * 3 + tid] = s;
  }
}

// Combine halves, average, L2-normalize over the 3 output dims.
__global__ void eff_attn_combine_kernel(const float* __restrict__ partial,
                                        float* __restrict__ out) {
  const int b = threadIdx.x + blockIdx.x * blockDim.x;
  if (b >= 256) return;
  const float a0 = 0.5f * (partial[(2 * b) * 3 + 0] + partial[(2 * b + 1) * 3 + 0]);
  const float a1 = 0.5f * (partial[(2 * b) * 3 + 1] + partial[(2 * b + 1) * 3 + 1]);
  const float a2 = 0.5f * (partial[(2 * b) * 3 + 2] + partial[(2 * b + 1) * 3 + 2]);
  const float inv = rsqrtf(a0 * a0 + a1 * a1 + a2 * a2);
  out[b * 3 + 0] = a0 * inv;
  out[b * 3 + 1] = a1 * inv;
  out[b * 3 + 2] = a2 * inv;
}

extern "C" void kernel_launch(void* const* d_in, const int* in_sizes, int n_in,
                              void* d_out, int out_size, void* d_ws, size_t ws_size,
                              hipStream_t stream) {
  (void)in_sizes; (void)n_in; (void)out_size; (void)ws_size;
  const float* x  = (const float*)d_in[0];
  const float* Wq = (const float*)d_in[1];
  const float* bq = (const float*)d_in[2];
  const float* Wk = (const float*)d_in[3];
  const float* bk = (const float*)d_in[4];
  const float* Wv = (const float*)d_in[5];
  const float* bv = (const float*)d_in[6];
  float* out = (float*)d_out;
  float* partial = (float*)d_ws;  // 512*3 floats

  eff_attn_half_kernel<<<512, 256, 0, stream>>>(x, Wq, bq, Wk, bk, Wv, bv, partial);
  eff_attn_combine_kernel<<<1, 256, 0, stream>>>(partial, out);
}